// MutationAttention_87462714016081
// MI455X (gfx1250) — compile-verified
//
#include <hip/hip_runtime.h>
#include <math.h>

typedef __attribute__((ext_vector_type(2))) float v2f;
typedef __attribute__((ext_vector_type(8))) float v8f;

#define NUM_GENES 4384
#define D_MODEL   128
#define BATCH     8
#define KTILES    (NUM_GENES / 16)              /* 274 */
#define KHALF     (KTILES / 2)                  /* 137 */
#define ROWTILES_PER_B (NUM_GENES / 16)         /* 274 */
#define TOTAL_ROWTILES (BATCH * ROWTILES_PER_B) /* 2192 */
#define VMIX_CHUNK (NUM_GENES / 8)              /* 548 */

// ---------------------------------------------------------------------------
// Kernel 1: Pp[hi][k] = scale * (W_in[2hi]·key[k], W_in[2hi+1]·key[k])
//           ec[k]     = exp(scale * b_in·key[k])   (bias folded multiplicatively)
// Also zero the w[b,k] column-sum accumulator and the mean buffer.
// ---------------------------------------------------------------------------
__global__ void prep_kernel(const float* __restrict__ W_in,
                            const float* __restrict__ b_in,
                            const float* __restrict__ key,
                            float2* __restrict__ Pp,
                            float* __restrict__ ec,
                            float* __restrict__ w_accum,
                            float* __restrict__ mean) {
    int k = blockIdx.x * blockDim.x + threadIdx.x;
    if (k < BATCH * D_MODEL) mean[k] = 0.f;
    if (k >= NUM_GENES) return;
    float a0 = 0.f, a1 = 0.f, a2 = 0.f, a3 = 0.f, ac = 0.f;
    const float* kr = key + (size_t)k * D_MODEL;
    #pragma unroll 4
    for (int d = 0; d < D_MODEL; ++d) {
        float kv = kr[d];
        a0 += W_in[0 * D_MODEL + d] * kv;
        a1 += W_in[1 * D_MODEL + d] * kv;
        a2 += W_in[2 * D_MODEL + d] * kv;
        a3 += W_in[3 * D_MODEL + d] * kv;
        ac += b_in[d] * kv;
    }
    const float scale = 0.0883883476483184f; /* 1/sqrt(128) */
    Pp[0 * NUM_GENES + k] = make_float2(a0 * scale, a1 * scale);
    Pp[1 * NUM_GENES + k] = make_float2(a2 * scale, a3 * scale);
    ec[k] = __expf(ac * scale);
    #pragma unroll
    for (int b = 0; b < BATCH; ++b) w_accum[b * NUM_GENES + k] = 0.f;
}

// ---------------------------------------------------------------------------
// Kernel 2: fused rank-4 score WMMA + softmax column sums.
// Wave pairs split the 274 k-tiles (137 each) for one 16-row q-tile; row sums
// are exchanged through LDS. Each tile = one V_WMMA_F32_16X16X4_F32 with
// inline-zero C; the score bias rides multiplicatively via ec[k].
// ---------------------------------------------------------------------------
__global__ void __launch_bounds__(256)
attn_kernel(const float* __restrict__ M,
            const float2* __restrict__ Pp,
            const float* __restrict__ ec,
            float* __restrict__ w_accum) {
    __shared__ float lsh[8][16];

    const int lane   = threadIdx.x & 31;
    const int wib    = threadIdx.x >> 5;   /* 0..7 */
    const int half   = wib & 1;            /* k-range half */
    const int qlocal = wib >> 1;           /* 0..3 q-tiles per block */
    const int gq     = blockIdx.x * 4 + qlocal;
    const int b      = gq / ROWTILES_PER_B;
    const int qt     = gq % ROWTILES_PER_B;
    const int lm     = lane & 15;
    const int hi     = lane >> 4;

    // A-matrix (16x4 f32): lanes 0-15 hold K=0,1 ; lanes 16-31 hold K=2,3.
    const float2* m2 = (const float2*)M;
    float2 av = m2[((size_t)b * NUM_GENES + qt * 16 + lm) * 2 + hi];
    v2f A; A.x = av.x; A.y = av.y;

    // B-matrix: one b64 load gives both K rows this lane contributes.
    const float2* Bbase = Pp + (size_t)hi * NUM_GENES;

    const int kt0 = half * KHALF;
    const int kt1 = kt0 + KHALF;

    float l[8];
    #pragma unroll
    for (int j = 0; j < 8; ++j) l[j] = 0.f;

    // ---- Pass A: row sums of exp(score)·ec (no max-shift: scores ~ N(0,1)) --
    for (int kt = kt0; kt < kt1; ++kt) {
        int k0 = kt * 16;
        float2 bv = Bbase[k0 + lm];
        v2f Bv; Bv.x = bv.x; Bv.y = bv.y;
        float e = ec[k0 + lm];
        v8f C = {};
        v8f D = __builtin_amdgcn_wmma_f32_16x16x4_f32(
            false, A, false, Bv, (short)0, C, false, false);
        #pragma unroll
        for (int j = 0; j < 8; ++j) l[j] += __expf(D[j]) * e;
    }
    // Reduce across the 16 lanes of each row.
    #pragma unroll
    for (int j = 0; j < 8; ++j) {
        float v = l[j];
        v += __shfl_xor(v, 1, 32);
        v += __shfl_xor(v, 2, 32);
        v += __shfl_xor(v, 4, 32);
        v += __shfl_xor(v, 8, 32);
        l[j] = v;
    }
    // Combine with partner wave's k-half through LDS.
    if (lm == 0) {
        #pragma unroll
        for (int j = 0; j < 8; ++j) lsh[wib][hi * 8 + j] = l[j];
    }
    __syncthreads();
    #pragma unroll
    for (int j = 0; j < 8; ++j)
        l[j] = 1.0f / (l[j] + lsh[wib ^ 1][hi * 8 + j]);

    // ---- Pass B: normalized probabilities -> column sums into w[b,k] -------
    float* wb = w_accum + (size_t)b * NUM_GENES;
    for (int kt = kt0; kt < kt1; ++kt) {
        int k0 = kt * 16;
        float2 bv = Bbase[k0 + lm];
        v2f Bv; Bv.x = bv.x; Bv.y = bv.y;
        float e = ec[k0 + lm];
        v8f C = {};
        v8f D = __builtin_amdgcn_wmma_f32_16x16x4_f32(
            false, A, false, Bv, (short)0, C, false, false);
        float col = 0.f;
        #pragma unroll
        for (int j = 0; j < 8; ++j) col += __expf(D[j]) * l[j];
        col *= e;                      /* ec[k] constant down a column */
        col += __shfl_xor(col, 16, 32);/* rows 0-7 + rows 8-15 */
        if (lane < 16) atomicAdd(&wb[k0 + lane], col);
    }
}

// ---------------------------------------------------------------------------
// Kernel 3: mean[b,d] += (1/G) * sum_{k in chunk} w[b,k] * V[k,d]
// ---------------------------------------------------------------------------
__global__ void vmix_kernel(const float* __restrict__ w_accum,
                            const float* __restrict__ V,
                            float* __restrict__ mean) {
    int b = blockIdx.x;
    int d = threadIdx.x; /* 128 */
    const float* wb = w_accum + (size_t)b * NUM_GENES;
    int k0 = blockIdx.y * VMIX_CHUNK;
    int k1 = k0 + VMIX_CHUNK;
    float acc = 0.f;
    #pragma unroll 4
    for (int k = k0; k < k1; ++k)
        acc += wb[k] * V[(size_t)k * D_MODEL + d];
    atomicAdd(&mean[b * D_MODEL + d], acc * (1.0f / (float)NUM_GENES));
}

// ---------------------------------------------------------------------------
// Kernel 4: out = GELU(mean@W1 + b1) @ W2 + b2   (exact erf GELU)
// ---------------------------------------------------------------------------
__global__ void mlp_kernel(const float* __restrict__ mean,
                           const float* __restrict__ W1,
                           const float* __restrict__ b1,
                           const float* __restrict__ W2,
                           const float* __restrict__ b2,
                           float* __restrict__ out) {
    __shared__ float h[64];
    int b = blockIdx.x;
    int t = threadIdx.x; /* 64 */
    float acc = b1[t];
    #pragma unroll 4
    for (int d = 0; d < D_MODEL; ++d)
        acc += mean[b * D_MODEL + d] * W1[d * 64 + t];
    h[t] = 0.5f * acc * (1.0f + erff(acc * 0.7071067811865475f));
    __syncthreads();
    if (t < 26) {
        float o = b2[t];
        #pragma unroll 4
        for (int j = 0; j < 64; ++j)
            o += h[j] * W2[j * 26 + t];
        out[b * 26 + t] = o;
    }
}

// ---------------------------------------------------------------------------
extern "C" void kernel_launch(void* const* d_in, const int* in_sizes, int n_in,
                              void* d_out, int out_size, void* d_ws, size_t ws_size,
                              hipStream_t stream) {
    const float* M    = (const float*)d_in[0]; // [8,4384,4]
    const float* V    = (const float*)d_in[1]; // [4384,128]
    const float* W_in = (const float*)d_in[2]; // [4,128]
    const float* b_in = (const float*)d_in[3]; // [128]
    const float* key  = (const float*)d_in[4]; // [4384,128]
    const float* W1   = (const float*)d_in[5]; // [128,64]
    const float* b1   = (const float*)d_in[6]; // [64]
    const float* W2   = (const float*)d_in[7]; // [64,26]
    const float* b2   = (const float*)d_in[8]; // [26]

    float*  ws   = (float*)d_ws;
    float2* Pp   = (float2*)ws;                        // 2*G float2 = 4*G floats
    float*  ec   = ws + 4 * NUM_GENES;                 // G
    float*  w    = ws + 5 * NUM_GENES;                 // B*G
    float*  mean = ws + (5 + BATCH) * NUM_GENES;       // B*128
    float*  out  = (float*)d_out;                      // [8,26]

    prep_kernel<<<(NUM_GENES + 255) / 256, 256, 0, stream>>>(
        W_in, b_in, key, Pp, ec, w, mean);
    attn_kernel<<<TOTAL_ROWTILES / 4, 256, 0, stream>>>(M, Pp, ec, w);
    vmix_kernel<<<dim3(BATCH, 8), D_MODEL, 0, stream>>>(w, V, mean);
    mlp_kernel<<<BATCH, 64, 0, stream>>>(mean, W1, b1, W2, b2, out);
}